// MultiScaleTriplane_Pooling_51342039056742
// MI455X (gfx1250) — compile-verified
//
#include <hip/hip_runtime.h>

// MI455X / gfx1250 implementation of MultiScaleTriplane_Pooling.
//
// Roofline: output write = 500000*64*4B = 128MB -> ~5.5us @ 23.3TB/s HBM.
// Processed planes = 3*128*128*64*4B = 12.6MB -> resident in 192MB L2, so the
// ~6GB of bicubic gather traffic is L2-bound. Conv stage is a 16384x64x576
// GEMM per plane (~3.6 GFLOP total) -> v_wmma_f32_16x16x4_f32 with 2x2
// register tiling (4 WMMAs per 8 operand loads).
//
// Workspace layout (floats), total ~25.2 MB:
//   [0)                ws_conv : 3*16384*64   conv+bias, HWC per plane
//   [3145728)          ws_plane: 3*16384*64   final sampling planes, HWC
//   [6291456)          sc_sh   : 3*64*2       folded BN scale/shift

#define MTOT   16384            // 128*128 output spatial positions
#define KTOT   576              // 64 in-channels * 9 taps
#define NPTS   500000
#define BN_EPS 1e-5f

typedef __attribute__((ext_vector_type(2))) float v2f;
typedef __attribute__((ext_vector_type(8))) float v8f;

// ---------------------------------------------------------------------------
// K1: conv3x3 stride2 pad1 + bias, via V_WMMA_F32_16X16X4_F32.
// GEMM view: C[m,n] = sum_k A[m,k]*B[k,n], m=spatial(128x128), n=oc, k=ic*9+tap.
// Wave tile: 32M x 32N (2x2 fragments); block = 8 waves = 128M x 64N.
// Fragment layouts per ISA 7.12.2:
//   A 16x4 : row M = lane%16 ; lane<16 holds K={0,1}, lane>=16 holds K={2,3}.
//   B 4x16 : col N = lane%16 ; lane<16 holds K={0,1}, lane>=16 holds K={2,3}.
//   C 16x16: VGPR r -> M = r + (lane<16 ? 0 : 8), N = lane%16.
// ---------------------------------------------------------------------------
__global__ void __launch_bounds__(256)
k1_conv_wmma(const float* __restrict__ in0, const float* __restrict__ in1,
             const float* __restrict__ in2, const float* __restrict__ conv_w,
             const float* __restrict__ conv_b, float* __restrict__ ws_conv)
{
    const int p = blockIdx.z;
    const float* __restrict__ pin = (p == 0) ? in0 : ((p == 1) ? in1 : in2);

    const int lane = threadIdx.x & 31;
    const int wave = threadIdx.x >> 5;
    const int l16  = lane & 15;
    const int hi   = lane >> 4;                 // 0: K pair {0,1}, 1: {2,3}

    const int mBase = blockIdx.x * 128 + (wave & 3) * 32;  // M tiles: mBase, mBase+16
    const int nBase = (wave >> 2) * 32;                    // N tiles: nBase, nBase+16

    const int mA0 = mBase + l16, mA1 = mA0 + 16;           // A rows this lane feeds
    const int oy0 = mA0 >> 7, ox0 = mA0 & 127;
    const int oy1 = mA1 >> 7, ox1 = mA1 & 127;
    const int nB0 = nBase + l16, nB1 = nB0 + 16;           // B cols this lane feeds
    const float* __restrict__ w0 = conv_w + nB0 * KTOT;
    const float* __restrict__ w1 = conv_w + nB1 * KTOT;

    v8f acc00 = {}, acc01 = {}, acc10 = {}, acc11 = {};

#pragma unroll 4
    for (int k0 = 0; k0 < KTOT; k0 += 4) {
        const int ka = k0 + hi * 2;
        v2f a0, a1, b0, b1;
#pragma unroll
        for (int u = 0; u < 2; ++u) {
            const int k  = ka + u;
            const int ic = k / 9, kk = k % 9;
            const int ky = kk / 3, kx = kk % 3;
            const int iy0 = 2 * oy0 - 1 + ky, ix0 = 2 * ox0 - 1 + kx;
            const int iy1 = 2 * oy1 - 1 + ky, ix1 = 2 * ox1 - 1 + kx;
            float v0 = 0.0f, v1 = 0.0f;
            if (((unsigned)iy0 < 256u) & ((unsigned)ix0 < 256u))
                v0 = pin[ic * 65536 + iy0 * 256 + ix0];
            if (((unsigned)iy1 < 256u) & ((unsigned)ix1 < 256u))
                v1 = pin[ic * 65536 + iy1 * 256 + ix1];
            a0[u] = v0; a1[u] = v1;
            b0[u] = w0[k]; b1[u] = w1[k];
        }
        // 4 WMMAs reuse the 2 A and 2 B fragments; EXEC all-ones on this path.
        acc00 = __builtin_amdgcn_wmma_f32_16x16x4_f32(false, a0, false, b0, (short)0, acc00, false, false);
        acc01 = __builtin_amdgcn_wmma_f32_16x16x4_f32(false, a0, false, b1, (short)0, acc01, false, false);
        acc10 = __builtin_amdgcn_wmma_f32_16x16x4_f32(false, a1, false, b0, (short)0, acc10, false, false);
        acc11 = __builtin_amdgcn_wmma_f32_16x16x4_f32(false, a1, false, b1, (short)0, acc11, false, false);
    }

    // Bias + store channel-last (HWC): 16 lanes write 64B contiguous runs.
    const float bias0 = conv_b[nB0], bias1 = conv_b[nB1];
#pragma unroll
    for (int r = 0; r < 8; ++r) {
        const int m0 = mBase + r + hi * 8;
        const int m1 = m0 + 16;
        ws_conv[((size_t)p * MTOT + m0) * 64 + nB0] = acc00[r] + bias0;
        ws_conv[((size_t)p * MTOT + m0) * 64 + nB1] = acc01[r] + bias1;
        ws_conv[((size_t)p * MTOT + m1) * 64 + nB0] = acc10[r] + bias0;
        ws_conv[((size_t)p * MTOT + m1) * 64 + nB1] = acc11[r] + bias1;
    }
}

// ---------------------------------------------------------------------------
// K2: deterministic BN statistics. One block per (plane, channel); LDS tree
// reduction of sum / sumsq over 16384 positions, then fold gamma/beta into
// a single scale/shift pair:  y = v*scale + shift.
// ---------------------------------------------------------------------------
__global__ void __launch_bounds__(256)
k2_stats(const float* __restrict__ ws_conv, const float* __restrict__ gamma,
         const float* __restrict__ beta, float* __restrict__ sc_sh)
{
    __shared__ float sh[256], sh2[256];
    const int pc = blockIdx.x;                         // p*64 + c, 0..191
    const int c  = pc & 63;
    const float* __restrict__ col = ws_conv + (size_t)(pc >> 6) * MTOT * 64 + c;

    float s = 0.0f, s2 = 0.0f;
    for (int m = threadIdx.x; m < MTOT; m += 256) {
        const float v = col[(size_t)m * 64];
        s += v; s2 += v * v;
    }
    sh[threadIdx.x] = s; sh2[threadIdx.x] = s2;
    __syncthreads();
    for (int off = 128; off > 0; off >>= 1) {
        if ((int)threadIdx.x < off) {
            sh[threadIdx.x]  += sh[threadIdx.x + off];
            sh2[threadIdx.x] += sh2[threadIdx.x + off];
        }
        __syncthreads();
    }
    if (threadIdx.x == 0) {
        const float mu    = sh[0] * (1.0f / (float)MTOT);
        const float var   = sh2[0] * (1.0f / (float)MTOT) - mu * mu;  // biased
        const float scale = rsqrtf(var + BN_EPS) * gamma[c];
        sc_sh[pc * 2 + 0] = scale;
        sc_sh[pc * 2 + 1] = beta[c] - mu * scale;
    }
}

// ---------------------------------------------------------------------------
// K3: ws_plane = relu(conv*scale + shift) + avgpool3x3_s2_p1(input)/9.
// block (64 ch, 4 x): conv read + final write are 256B-coalesced in HWC.
// ---------------------------------------------------------------------------
__global__ void __launch_bounds__(256)
k3_finalize(const float* __restrict__ in0, const float* __restrict__ in1,
            const float* __restrict__ in2, const float* __restrict__ ws_conv,
            const float* __restrict__ sc_sh, float* __restrict__ ws_plane)
{
    const int c = threadIdx.x;                         // 0..63
    const int x = blockIdx.x * 4 + threadIdx.y;        // 0..127
    const int y = blockIdx.y;                          // 0..127
    const int p = blockIdx.z;
    const float* __restrict__ pin = (p == 0) ? in0 : ((p == 1) ? in1 : in2);

    const size_t idx = ((size_t)p * MTOT + y * 128 + x) * 64 + c;
    float v = ws_conv[idx];
    v = fmaxf(v * sc_sh[(p * 64 + c) * 2] + sc_sh[(p * 64 + c) * 2 + 1], 0.0f);

    // avgpool3x3 stride2 pad1, count_include_pad (always /9, zeros at pad)
    float pool = 0.0f;
    const float* __restrict__ pc = pin + c * 65536;
#pragma unroll
    for (int dy = 0; dy < 3; ++dy) {
        const int iy = 2 * y - 1 + dy;
        if ((unsigned)iy >= 256u) continue;
#pragma unroll
        for (int dx = 0; dx < 3; ++dx) {
            const int ix = 2 * x - 1 + dx;
            if ((unsigned)ix < 256u) pool += pc[iy * 256 + ix];
        }
    }
    ws_plane[idx] = v + pool * (1.0f / 9.0f);
}

// ---------------------------------------------------------------------------
// K4: bicubic sampling (border clamp, align_corners, torch a=-0.75).
// One wave32 per point; lane owns channel pair {2*lane, 2*lane+1}. Each of the
// 48 taps is one coalesced 256B row read (float2/lane) out of L2-resident
// planes. Row bases are prefetched (global_prefetch_b8) while the cubic
// weights are computed. Output store is one contiguous 256B float2 store.
// ---------------------------------------------------------------------------
__device__ __forceinline__ void cubic_w(float t, float w[4])
{
    const float a = -0.75f;
    const float t1 = t + 1.0f;
    w[0] = ((a * t1 - 5.0f * a) * t1 + 8.0f * a) * t1 - 4.0f * a;
    w[1] = ((a + 2.0f) * t - (a + 3.0f)) * t * t + 1.0f;
    const float s = 1.0f - t;
    w[2] = ((a + 2.0f) * s - (a + 3.0f)) * s * s + 1.0f;
    w[3] = 1.0f - w[0] - w[1] - w[2];
}

__global__ void __launch_bounds__(256)
k4_sample(const float* __restrict__ coords, const float* __restrict__ noise,
          const float* __restrict__ planes, float* __restrict__ out)
{
    const int wave = threadIdx.x >> 5;
    const int lane = threadIdx.x & 31;
    const int pt   = blockIdx.x * 8 + wave;
    if (pt >= NPTS) return;                            // never taken (500000 = 62500*8)

    const float cx = (coords[pt * 3 + 0] + 1.0f) * 0.5f;
    const float cy = (coords[pt * 3 + 1] + 1.0f) * 0.5f;
    const float cz = (coords[pt * 3 + 2] + 1.0f) * 0.5f;

    float2 acc; acc.x = 0.0f; acc.y = 0.0f;

#pragma unroll
    for (int p = 0; p < 3; ++p) {
        // plane 0: (x,y)  plane 1: (y,z)  plane 2: (x,z); grid[:,0]->W, grid[:,1]->H
        float u = (p == 0) ? cx : ((p == 1) ? cy : cx);
        float v = (p == 0) ? cy : ((p == 1) ? cz : cz);
        u += noise[((size_t)p * NPTS + pt) * 2 + 0];
        v += noise[((size_t)p * NPTS + pt) * 2 + 1];

        const float gx = fminf(fmaxf((u + 1.0f) * 0.5f * 127.0f, 0.0f), 127.0f);
        const float gy = fminf(fmaxf((v + 1.0f) * 0.5f * 127.0f, 0.0f), 127.0f);
        const float x0f = floorf(gx), y0f = floorf(gy);
        const int   x0 = (int)x0f,    y0 = (int)y0f;

        const float* __restrict__ pb = planes + (size_t)p * MTOT * 64 + lane * 2;

        // Pull the 4 tap rows toward the WGP while VALU does the weight math.
        {
            const int xs = min(max(x0 - 1, 0), 127);
#pragma unroll
            for (int i = 0; i < 4; ++i) {
                const int yi = min(max(y0 + i - 1, 0), 127);
                __builtin_prefetch(pb + (size_t)yi * (128 * 64) + xs * 64, 0, 0);
            }
        }

        float wx[4], wy[4];
        cubic_w(gx - x0f, wx);
        cubic_w(gy - y0f, wy);

#pragma unroll
        for (int i = 0; i < 4; ++i) {
            const int yi = min(max(y0 + i - 1, 0), 127);
            const float* __restrict__ prow = pb + (size_t)yi * (128 * 64);
            const float wyi = wy[i];
#pragma unroll
            for (int j = 0; j < 4; ++j) {
                const int xi = min(max(x0 + j - 1, 0), 127);
                const float w = wyi * wx[j];
                const float2 val = *(const float2*)(prow + xi * 64);
                acc.x += w * val.x;
                acc.y += w * val.y;
            }
        }
    }
    *(float2*)(out + (size_t)pt * 64 + lane * 2) = acc;
}

// ---------------------------------------------------------------------------
extern "C" void kernel_launch(void* const* d_in, const int* in_sizes, int n_in,
                              void* d_out, int out_size, void* d_ws, size_t ws_size,
                              hipStream_t stream)
{
    const float* coords = (const float*)d_in[0];   // (N,3)
    const float* noise  = (const float*)d_in[1];   // (3,N,2)
    const float* px     = (const float*)d_in[2];   // (1,64,256,256)
    const float* py     = (const float*)d_in[3];
    const float* pz     = (const float*)d_in[4];
    const float* cw     = (const float*)d_in[5];   // (64,64,3,3)
    const float* cb     = (const float*)d_in[6];   // (64,)
    const float* gamma  = (const float*)d_in[7];
    const float* beta   = (const float*)d_in[8];
    float*       out    = (float*)d_out;           // (N,64) fp32

    float* ws       = (float*)d_ws;                // needs ~25.2 MB
    float* ws_conv  = ws;                          // 3*16384*64
    float* ws_plane = ws + (size_t)3 * MTOT * 64;  // 3*16384*64
    float* sc_sh    = ws + (size_t)6 * MTOT * 64;  // 3*64*2

    // conv GEMM: 128 blocks of 128M x 64N per plane, 8 waves/block (2x2 tiles each)
    k1_conv_wmma<<<dim3(128, 1, 3), 256, 0, stream>>>(px, py, pz, cw, cb, ws_conv);
    // BN stats: one block per (plane, channel), deterministic LDS reduction
    k2_stats<<<192, 256, 0, stream>>>(ws_conv, gamma, beta, sc_sh);
    // BN + relu + avgpool add -> sampling planes (HWC, L2-resident)
    k3_finalize<<<dim3(32, 128, 3), dim3(64, 4, 1), 0, stream>>>(px, py, pz, ws_conv,
                                                                 sc_sh, ws_plane);
    // bicubic triplane gather: 1 wave per point
    k4_sample<<<62500, 256, 0, stream>>>(coords, noise, ws_plane, out);

    (void)in_sizes; (void)n_in; (void)out_size; (void)ws_size;
}